// CA_49701361549823
// MI455X (gfx1250) — compile-verified
//
#include <hip/hip_runtime.h>
#include <hip/hip_bf16.h>

// ---------------------------------------------------------------------------
// Squeeze-Excite for x[32,64,256,256] f32 on MI455X (gfx1250).
// Bandwidth-bound: ~1.5 GiB HBM traffic -> ~69us floor @ 23.3 TB/s.
// - Streaming passes use float4 (global_load/store_b128) with non-temporal
//   hints (512MiB tensor > 192MB L2: no reuse possible, don't pollute L2).
// - Reduction is a fixed-shape two-stage tree (deterministic, no f32 atomics).
// - The 64->4->64 channel MLP is done with V_WMMA_F32_16X16X4_F32 in a single
//   wave32 (EXEC all-1s as WMMA requires).
// ---------------------------------------------------------------------------

typedef __attribute__((ext_vector_type(2))) float v2f;
typedef __attribute__((ext_vector_type(4))) float f4;
typedef __attribute__((ext_vector_type(8))) float v8f;

#define NB    32          // batch
#define NC    64          // channels
#define NCH   (NB * NC)   // 2048 (b,c) channels
#define HW    65536       // 256*256 spatial elements per channel
#define SPL   16          // reduction splits per channel
#define CHUNK 4096        // floats per block in streaming kernels (HW/SPL)

// ---------------------------------------------------------------------------
// Stage 1: partial[ch][sp] = sum of x[ch*HW + sp*CHUNK .. +CHUNK)
// grid = NCH*SPL = 32768 blocks x 256 threads; each thread 4 float4 loads.
// ---------------------------------------------------------------------------
__global__ void se_reduce_kernel(const float* __restrict__ x,
                                 float* __restrict__ partial) {
    __shared__ float sdata[256];
    const int ch = blockIdx.x >> 4;   // / SPL
    const int sp = blockIdx.x & 15;   // % SPL
    const int t  = threadIdx.x;

    const f4* b4 = (const f4*)(x + (size_t)ch * HW + (size_t)sp * CHUNK);
    float s = 0.f;
#pragma unroll
    for (int i = 0; i < 4; ++i) {
        f4 v = __builtin_nontemporal_load(&b4[t + i * 256]);
        s += (v.x + v.y) + (v.z + v.w);
    }
    sdata[t] = s;
    __syncthreads();
#pragma unroll
    for (int off = 128; off > 0; off >>= 1) {
        if (t < off) sdata[t] += sdata[t + off];
        __syncthreads();
    }
    if (t == 0) partial[ch * SPL + sp] = sdata[0];
}

// ---------------------------------------------------------------------------
// Stage 2: one wave32 does pooled -> MLP -> scale using WMMA f32 16x16x4.
//
// VGPR layout assumptions (CDNA5 ISA 7.12.2, 32-bit operands, wave32):
//   A (16x4 MxK):  lane L holds M = L%16;  K = vgpr + (L<16 ? 0 : 2)
//   B (4x16 KxN):  lane L holds N = L%16;  K = vgpr + (L<16 ? 0 : 2)
//   C/D (16x16):   vgpr r, lane L -> M = r + (L<16 ? 0 : 8), N = L%16
// ---------------------------------------------------------------------------
__global__ void se_mlp_kernel(const float* __restrict__ partial, // [NCH][SPL]
                              const float* __restrict__ w_down,  // [4][64]
                              const float* __restrict__ b_down,  // [4]
                              const float* __restrict__ w_up,    // [64][4]
                              const float* __restrict__ b_up,    // [64]
                              float* __restrict__ scale) {       // [NCH]
    __shared__ float pooled[NB][NC];
    __shared__ float hidden[NB][4];

    const int lane  = threadIdx.x;        // 0..31
    const int nlane = lane & 15;          // N (or M) position within tile
    const int koff  = (lane < 16) ? 0 : 2;

    // Fold partials: lane 'b' computes pooled[b][c] for all 64 channels.
    // partial rows are 16 contiguous floats -> 4 f4 loads each.
    {
        const f4* p4 = (const f4*)(partial + (size_t)lane * NC * SPL);
        for (int c = 0; c < NC; ++c) {
            f4 a = p4[c * 4 + 0];
            f4 b = p4[c * 4 + 1];
            f4 cc = p4[c * 4 + 2];
            f4 d = p4[c * 4 + 3];
            float s = ((a.x + a.y) + (a.z + a.w)) + ((b.x + b.y) + (b.z + b.w)) +
                      ((cc.x + cc.y) + (cc.z + cc.w)) + ((d.x + d.y) + (d.z + d.w));
            pooled[lane][c] = s * (1.0f / 65536.0f);
        }
    }
    __syncthreads();

    // GEMM1: hidden[32][4] = relu(pooled[32][64] * w_down^T[64][4] + b_down)
    // Two 16-row M tiles, 16 K-steps of 4. N tile is 16 wide; cols 4..15 = 0.
#pragma unroll
    for (int mt = 0; mt < 2; ++mt) {
        v8f acc = {};
        const int m = mt * 16 + nlane;
        for (int kb = 0; kb < 16; ++kb) {
            const int k0 = kb * 4 + koff;
            v2f a, b;
            a.x = pooled[m][k0];
            a.y = pooled[m][k0 + 1];
            b.x = (nlane < 4) ? w_down[nlane * 64 + k0] : 0.f;
            b.y = (nlane < 4) ? w_down[nlane * 64 + k0 + 1] : 0.f;
            acc = __builtin_amdgcn_wmma_f32_16x16x4_f32(
                false, a, false, b, (short)0, acc, false, false);
        }
        if (nlane < 4) {
            const int mbase = mt * 16 + ((lane < 16) ? 0 : 8);
            const float bd = b_down[nlane];
#pragma unroll
            for (int r = 0; r < 8; ++r) {
                float h = acc[r] + bd;
                hidden[mbase + r][nlane] = fmaxf(h, 0.f);
            }
        }
    }
    __syncthreads();

    // GEMM2: scale[32][64] = sigmoid(hidden[32][4] * w_up^T[4][64] + b_up)
    // Two M tiles x four N tiles, single K-step of 4.
#pragma unroll
    for (int mt = 0; mt < 2; ++mt) {
        v2f a;
        const int m = mt * 16 + nlane;
        a.x = hidden[m][koff];
        a.y = hidden[m][koff + 1];
#pragma unroll
        for (int nt = 0; nt < 4; ++nt) {
            const int c = nt * 16 + nlane;
            v2f b;
            b.x = w_up[c * 4 + koff];
            b.y = w_up[c * 4 + koff + 1];
            v8f acc = {};
            acc = __builtin_amdgcn_wmma_f32_16x16x4_f32(
                false, a, false, b, (short)0, acc, false, false);
            const int mbase = mt * 16 + ((lane < 16) ? 0 : 8);
            const float bu = b_up[c];
#pragma unroll
            for (int r = 0; r < 8; ++r) {
                float v = acc[r] + bu;
                v = 1.0f / (1.0f + __expf(-v));
                scale[(mbase + r) * NC + c] = v;
            }
        }
    }
}

// ---------------------------------------------------------------------------
// Stage 3: out = x * scale[ch], broadcast per channel.
// Each block owns 4096 contiguous floats (always inside one channel since
// 4096 | 65536). grid = 32768 blocks x 256 threads, 4 float4 per thread, NT.
// ---------------------------------------------------------------------------
__global__ void se_scale_kernel(const float* __restrict__ x,
                                const float* __restrict__ scale,
                                float* __restrict__ out) {
    const int ch  = blockIdx.x >> 4;          // 16 blocks per channel
    const float s = scale[ch];                 // scalar (uniform per block)
    const size_t base4 = (size_t)blockIdx.x * (CHUNK / 4);
    const f4* x4 = (const f4*)x;
    f4* o4 = (f4*)out;
    const int t = threadIdx.x;
#pragma unroll
    for (int i = 0; i < 4; ++i) {
        const size_t g = base4 + (size_t)i * 256 + t;
        f4 v = __builtin_nontemporal_load(&x4[g]);
        v.x *= s; v.y *= s; v.z *= s; v.w *= s;
        __builtin_nontemporal_store(v, &o4[g]);
    }
}

// ---------------------------------------------------------------------------
extern "C" void kernel_launch(void* const* d_in, const int* in_sizes, int n_in,
                              void* d_out, int out_size, void* d_ws, size_t ws_size,
                              hipStream_t stream) {
    const float* trans_b = (const float*)d_in[0];  // [32,64,256,256]
    const float* w_down  = (const float*)d_in[1];  // [4,64]
    const float* b_down  = (const float*)d_in[2];  // [4]
    const float* w_up    = (const float*)d_in[3];  // [64,4]
    const float* b_up    = (const float*)d_in[4];  // [64]
    float* out = (float*)d_out;

    float* partial = (float*)d_ws;            // NCH*SPL = 32768 floats (128 KiB)
    float* scale   = partial + NCH * SPL;     // NCH = 2048 floats

    se_reduce_kernel<<<NCH * SPL, 256, 0, stream>>>(trans_b, partial);
    se_mlp_kernel<<<1, 32, 0, stream>>>(partial, w_down, b_down, w_up, b_up, scale);
    se_scale_kernel<<<NCH * SPL, 256, 0, stream>>>(trans_b, scale, out);
}